// CausalTrajectoryPrediction_28698971471877
// MI455X (gfx1250) — compile-verified
//
#include <hip/hip_runtime.h>

#define N_IN 512
#define HID  256
#define MED  64

typedef __attribute__((ext_vector_type(2))) float v2f;
typedef __attribute__((ext_vector_type(8))) float v8f;

__device__ __forceinline__ float relu(float v) { return v > 0.f ? v : 0.f; }

__device__ __forceinline__ v8f wmma_k4(v2f a, v2f b, v8f c) {
    // D = A(16x4 f32) * B(4x16 f32) + C(16x16 f32)
    return __builtin_amdgcn_wmma_f32_16x16x4_f32(
        /*neg_a=*/false, a, /*neg_b=*/false, b,
        /*c_mod=*/(short)0, c, /*reuse_a=*/false, /*reuse_b=*/false);
}

__global__ __launch_bounds__(256) void ctp_kernel(
    const float* __restrict__ x,
    const float* __restrict__ W1a,
    const float* __restrict__ W1b,
    const float* __restrict__ W2a,
    const float* __restrict__ b2a,
    const float* __restrict__ W2b,
    const float* __restrict__ b2b,
    const float* __restrict__ W4a,
    const float* __restrict__ b4a,
    const float* __restrict__ W4b,
    const float* __restrict__ b4b,
    const float* __restrict__ noise,
    float* __restrict__ out)
{
    __shared__ float xs[N_IN];    // x with element n zeroed
    __shared__ float h1s[HID];
    __shared__ float r3s[68];     // r1[0..63], r2 at [64], zero pad [65..67]
    __shared__ float h4s[HID];
    __shared__ float red0[256];
    __shared__ float red1[256];

    const int n    = blockIdx.x;
    const int t    = threadIdx.x;
    const int wave = t >> 5;
    const int lane = t & 31;
    const int lrow = lane & 15;          // row within 16-row tile
    const int koff = (lane >> 4) << 1;   // 0 for lanes 0-15, 2 for lanes 16-31

    // Stage 0: masked input vector (x_off)
    xs[t]       = (t == n)       ? 0.f : x[t];
    xs[t + 256] = (t + 256 == n) ? 0.f : x[t + 256];
    if (t < 3) r3s[65 + t] = 0.f;
    __syncthreads();

    // ---- Stage 1: h1 = relu(W1a[n] @ x_off)  (256x512 matvec) ----
    {
        const float* Wn = W1a + (size_t)n * HID * N_IN;
        #pragma unroll
        for (int tile = 0; tile < 2; ++tile) {
            const int m0 = tile * 128 + wave * 16;
            const float* Wrow = Wn + (size_t)(m0 + lrow) * N_IN;
            v8f c = {};
            #pragma unroll 4
            for (int k = 0; k < N_IN; k += 4) {
                v2f a = __builtin_nontemporal_load((const v2f*)(Wrow + k + koff));
                v2f b;
                b.x = xs[k + koff];
                b.y = xs[k + koff + 1];
                c = wmma_k4(a, b, c);
            }
            if (lane == 0) {
                #pragma unroll
                for (int v = 0; v < 8; ++v) h1s[m0 + v] = relu(c[v]);
            }
            if (lane == 16) {
                #pragma unroll
                for (int v = 0; v < 8; ++v) h1s[m0 + 8 + v] = relu(c[v]);
            }
        }
    }
    __syncthreads();

    // ---- Stage 2: r1 = relu(W1b[n] @ h1)  (64x256), waves 0-3 ----
    if (wave < 4) {
        const float* Wn = W1b + (size_t)n * MED * HID;
        const int m0 = wave * 16;
        const float* Wrow = Wn + (size_t)(m0 + lrow) * HID;
        v8f c = {};
        #pragma unroll 4
        for (int k = 0; k < HID; k += 4) {
            v2f a = __builtin_nontemporal_load((const v2f*)(Wrow + k + koff));
            v2f b;
            b.x = h1s[k + koff];
            b.y = h1s[k + koff + 1];
            c = wmma_k4(a, b, c);
        }
        if (lane == 0) {
            #pragma unroll
            for (int v = 0; v < 8; ++v) r3s[m0 + v] = relu(c[v]);
        }
        if (lane == 16) {
            #pragma unroll
            for (int v = 0; v < 8; ++v) r3s[m0 + 8 + v] = relu(c[v]);
        }
    }

    // ---- Stage 3: self-excite net (x_diag has one nonzero -> 2 scalars) ----
    if (t == 255) {
        float xn  = x[n];
        float h20 = relu(W2a[(size_t)n * 2 * N_IN + 0 * N_IN + n] * xn + b2a[n * 2 + 0]);
        float h21 = relu(W2a[(size_t)n * 2 * N_IN + 1 * N_IN + n] * xn + b2a[n * 2 + 1]);
        float r2  = relu(W2b[n * 2 + 0] * h20 + W2b[n * 2 + 1] * h21 + b2b[n]);
        r3s[64] = r2;
    }
    __syncthreads();

    // ---- Stage 4: h4 = relu(W4a[n] @ r3 + b4a)  (256x65, K padded to 68) ----
    {
        const float* Wn = W4a + (size_t)n * HID * 65;
        const float* b4an = b4a + (size_t)n * HID;
        #pragma unroll
        for (int tile = 0; tile < 2; ++tile) {
            const int m0 = tile * 128 + wave * 16;
            const float* Wrow = Wn + (size_t)(m0 + lrow) * 65;
            v8f c = {};
            #pragma unroll
            for (int k = 0; k < 68; k += 4) {
                const int kk = k + koff;
                v2f a;
                a.x = (kk     < 65) ? Wrow[kk]     : 0.f;   // guard: odd stride + tail
                a.y = (kk + 1 < 65) ? Wrow[kk + 1] : 0.f;
                v2f b;
                b.x = r3s[kk];
                b.y = r3s[kk + 1];
                c = wmma_k4(a, b, c);
            }
            if (lane == 0) {
                #pragma unroll
                for (int v = 0; v < 8; ++v) {
                    int r = m0 + v;
                    h4s[r] = relu(c[v] + b4an[r]);
                }
            }
            if (lane == 16) {
                #pragma unroll
                for (int v = 0; v < 8; ++v) {
                    int r = m0 + 8 + v;
                    h4s[r] = relu(c[v] + b4an[r]);
                }
            }
        }
    }
    __syncthreads();

    // ---- Stage 5: pred = relu(W4b[n] @ h4 + b4b); deterministic tree reduce ----
    {
        float h = h4s[t];
        red0[t] = __builtin_nontemporal_load(W4b + (size_t)n * 2 * HID + t) * h;
        red1[t] = __builtin_nontemporal_load(W4b + (size_t)n * 2 * HID + HID + t) * h;
    }
    __syncthreads();
    #pragma unroll
    for (int s = 128; s > 0; s >>= 1) {
        if (t < s) {
            red0[t] += red0[t + s];
            red1[t] += red1[t + s];
        }
        __syncthreads();
    }
    if (t == 0) {
        float p0 = relu(red0[0] + b4b[n * 2 + 0]);
        float p1 = relu(red1[0] + b4b[n * 2 + 1]);
        out[n] = p0 + noise[n] * p1;
    }
}

extern "C" void kernel_launch(void* const* d_in, const int* in_sizes, int n_in,
                              void* d_out, int out_size, void* d_ws, size_t ws_size,
                              hipStream_t stream) {
    (void)in_sizes; (void)n_in; (void)d_ws; (void)ws_size; (void)out_size;
    const float* x    = (const float*)d_in[0];
    const float* W1a  = (const float*)d_in[1];
    const float* W1b  = (const float*)d_in[2];
    const float* W2a  = (const float*)d_in[3];
    const float* b2a  = (const float*)d_in[4];
    const float* W2b  = (const float*)d_in[5];
    const float* b2b  = (const float*)d_in[6];
    const float* W4a  = (const float*)d_in[7];
    const float* b4a  = (const float*)d_in[8];
    const float* W4b  = (const float*)d_in[9];
    const float* b4b  = (const float*)d_in[10];
    const float* nois = (const float*)d_in[11];
    float* out = (float*)d_out;

    ctp_kernel<<<N_IN, 256, 0, stream>>>(x, W1a, W1b, W2a, b2a, W2b, b2b,
                                         W4a, b4a, W4b, b4b, nois, out);
}